// YOLOV2Loss_56788057587769
// MI455X (gfx1250) — compile-verified
//
#include <hip/hip_runtime.h>

typedef __attribute__((ext_vector_type(2))) float v2f;
typedef __attribute__((ext_vector_type(8))) float v8f;

#define AN      5
#define KCH     85
#define CCH     425            // AN*KCH
#define HH      52
#define WW      52
#define HWP     (HH*WW)        // 2704
#define CHWP    (CCH*HWP)
#define NBATCH  64
#define NCELLS  (NBATCH*HWP)   // 173056, divisible by 256
#define THREADS 256
#define NBLOCKS (NCELLS/THREADS) // 676

__device__ __forceinline__ float ldnt(const float* p) {
    // streaming read-once data: non-temporal hint, don't thrash 192MB L2
    return __builtin_nontemporal_load(p);
}

__global__ __launch_bounds__(THREADS)
void yolo_loss_partial(const float* __restrict__ out,
                       const float* __restrict__ gt,
                       float* __restrict__ partial)
{
    const int tid  = threadIdx.x;
    const int cell = blockIdx.x * THREADS + tid;   // w-fastest -> coalesced
    const int n    = cell / HWP;
    const int hw   = cell - n * HWP;
    const float* ob = out + (size_t)n * CHWP + hw;
    const float* gb = gt  + (size_t)n * CHWP + hw;

    float acc = 0.0f;

    // gt anchor-0 box (IoU reference box for this cell)
    const float bx1 = ldnt(gb + 0*HWP);
    const float by1 = ldnt(gb + 1*HWP);
    const float bx2 = ldnt(gb + 2*HWP);
    const float by2 = ldnt(gb + 3*HWP);
    const float areaB = fmaxf(bx2 - bx1, 0.f) * fmaxf(by2 - by1, 0.f);

    float iou[AN], possq[AN], oconf[AN], hasobj[AN], noobj[AN];
#pragma unroll
    for (int a = 0; a < AN; ++a) {
        const float* oa = ob + a * KCH * HWP;
        const float* ga = gb + a * KCH * HWP;
        const float ox1 = ldnt(oa + 0*HWP), oy1 = ldnt(oa + 1*HWP);
        const float ox2 = ldnt(oa + 2*HWP), oy2 = ldnt(oa + 3*HWP);
        const float gx1 = ldnt(ga + 0*HWP), gy1 = ldnt(ga + 1*HWP);
        const float gx2 = ldnt(ga + 2*HWP), gy2 = ldnt(ga + 3*HWP);

        const float d0 = ox1-gx1, d1 = oy1-gy1, d2 = ox2-gx2, d3 = oy2-gy2;
        possq[a] = d0*d0 + d1*d1 + d2*d2 + d3*d3;

        const float ix1 = fmaxf(ox1, bx1), iy1 = fmaxf(oy1, by1);
        const float ix2 = fminf(ox2, bx2), iy2 = fminf(oy2, by2);
        const float iw  = fmaxf(ix2 - ix1, 0.f), ih = fmaxf(iy2 - iy1, 0.f);
        const float inter = iw * ih;
        const float areaA = fmaxf(ox2 - ox1, 0.f) * fmaxf(oy2 - oy1, 0.f);
        iou[a] = inter / (areaA + areaB - inter + 1e-8f);

        oconf[a] = ldnt(oa + 4*HWP);
        const float gc = ldnt(ga + 4*HWP);
        hasobj[a] = (gc > 0.f)  ? 1.f : 0.f;
        noobj[a]  = (gc == 0.f) ? 1.f : 0.f;
    }

    // first-max argmax + select (compile-time indices only -> stays in VGPRs)
    float bi = iou[0], bp = possq[0], bc = oconf[0], bh = hasobj[0];
#pragma unroll
    for (int a = 1; a < AN; ++a) {
        const bool g = iou[a] > bi;
        bi = g ? iou[a]    : bi;
        bp = g ? possq[a]  : bp;
        bc = g ? oconf[a]  : bc;
        bh = g ? hasobj[a] : bh;
    }
    acc += 5.0f * bp * bh;                 // W_POS * position loss
    const float dc = bc - bi;              // conf target = iou (stop-grad)
    acc += dc * dc * bh;                   // obj conf loss
#pragma unroll
    for (int a = 0; a < AN; ++a)           // W_NOOBJ * no-obj conf loss
        acc += 0.5f * noobj[a] * oconf[a] * oconf[a];

    // class-score loss: 2*400 streamed reads per cell (dominant traffic)
#pragma unroll
    for (int a = 0; a < AN; ++a) {
        const float* os = ob + (a*KCH + 5) * HWP;
        const float* gs = gb + (a*KCH + 5) * HWP;
        float s = 0.f;
#pragma unroll 8
        for (int j = 0; j < KCH - 5; ++j) {
            const float d = ldnt(os + j*HWP) - ldnt(gs + j*HWP);
            s = fmaf(d, d, s);
        }
        acc += hasobj[a] * s;
    }

    // ---- block reduction: 256 -> 32 via LDS, 32 -> 1 via one WMMA ----
    __shared__ float red[THREADS];
    red[tid] = acc;
    __syncthreads();
    if (tid < 32) {                        // wave 0 only: EXEC all-1s in-wave
        float p = 0.f;
#pragma unroll
        for (int i = 0; i < THREADS/32; ++i) p += red[tid + 32*i];

        // V_WMMA_F32_16X16X4_F32 row-sum trick:
        // A[m][k] = p (dup over K per lane), B = ones  =>  D[m][n] = 2*(p[m]+p[m+16])
        v2f aop; aop[0] = p;   aop[1] = p;
        v2f bop; bop[0] = 1.f; bop[1] = 1.f;
        v8f cop = {};
        v8f d = __builtin_amdgcn_wmma_f32_16x16x4_f32(
            /*neg_a=*/false, aop, /*neg_b=*/false, bop,
            /*c_mod=*/(short)0, cop, /*reuse_a=*/false, /*reuse_b=*/false);

        float s = 0.f;
#pragma unroll
        for (int r = 0; r < 8; ++r) s += d[r];   // lane0: rows 0-7 of col 0
        const float s_hi = __shfl(s, 16, 32);    // lane16: rows 8-15 of col 0
        if (tid == 0) partial[blockIdx.x] = (s + s_hi) * 0.5f;
    }
}

__global__ __launch_bounds__(THREADS)
void yolo_loss_finish(const float* __restrict__ partial, float* __restrict__ o)
{
    __shared__ float red[THREADS];
    float s = 0.f;
    for (int i = threadIdx.x; i < NBLOCKS; i += THREADS) s += partial[i];
    red[threadIdx.x] = s;
    __syncthreads();
    for (int off = THREADS/2; off > 0; off >>= 1) {
        if (threadIdx.x < off) red[threadIdx.x] += red[threadIdx.x + off];
        __syncthreads();
    }
    if (threadIdx.x == 0) o[0] = red[0] * (1.0f / NBATCH);
}

extern "C" void kernel_launch(void* const* d_in, const int* in_sizes, int n_in,
                              void* d_out, int out_size, void* d_ws, size_t ws_size,
                              hipStream_t stream)
{
    const float* out_t = (const float*)d_in[0];
    const float* gt_t  = (const float*)d_in[1];
    float* partial = (float*)d_ws;                 // NBLOCKS floats of scratch

    yolo_loss_partial<<<NBLOCKS, THREADS, 0, stream>>>(out_t, gt_t, partial);
    yolo_loss_finish<<<1, THREADS, 0, stream>>>(partial, (float*)d_out);
}